// MecanumSystem_89498528514423
// MI455X (gfx1250) — compile-verified
//
#include <hip/hip_runtime.h>
#include <stdint.h>

#ifndef __has_builtin
#define __has_builtin(x) 0
#endif

#define AS1 __attribute__((address_space(1)))
#define AS3 __attribute__((address_space(3)))

typedef int v4i __attribute__((vector_size(16)));   // b128 payload type

// ---------------- constants (folded in double, stored as float, matching
// the reference's numpy-float64 -> float32 constant pipeline) --------------
constexpr double dPI      = 3.14159265358979323846;
constexpr double dT_STALL = 5.4 / 100.0;
constexpr double dW_FREE  = 1620.0 / 60.0 * 2.0 * dPI;
constexpr double dA       = 0.129907 + 0.095724;
constexpr double dR       = 4.0 * 0.0254;
constexpr double dINERT   = 6.0;
constexpr double dMOI     = 6.0 * (12.0 * 0.0254) * (12.0 * 0.0254) / 6.0;

constexpr float F_T_STALL = (float)dT_STALL;
constexpr float F_INV_WF  = (float)(1.0 / dW_FREE);
constexpr float F_INV_R   = (float)(1.0 / dR);
constexpr float F_AR      = (float)(dA / dR);             // A/R term of LV2WV
constexpr float F_K1      = (float)(1.0 / (dR * dINERT)); // WT2LA xy rows
constexpr float F_K2      = (float)(dA / (dR * dMOI));    // WT2LA omega row

// ---------------- CDNA5 async global<->LDS helpers ------------------------
#if __has_builtin(__builtin_amdgcn_global_load_async_to_lds_b128) && \
    __has_builtin(__builtin_amdgcn_global_store_async_from_lds_b128)
#define HAS_ASYNC_BI 1
#endif

__device__ __forceinline__ void async_load_b128(const float4* g, float4* l) {
#ifdef HAS_ASYNC_BI
  __builtin_amdgcn_global_load_async_to_lds_b128((AS1 v4i*)g, (AS3 v4i*)l, 0, 0);
#else
  unsigned loff = (unsigned)(uintptr_t)(AS3 void*)l;           // LDS byte offset
  unsigned long long ga = (unsigned long long)(uintptr_t)g;    // 64-bit global
  asm volatile("global_load_async_to_lds_b128 %0, %1, off"
               :: "v"(loff), "v"(ga) : "memory");
#endif
}

__device__ __forceinline__ void async_store_b128(float4* g, const float4* l) {
#ifdef HAS_ASYNC_BI
  __builtin_amdgcn_global_store_async_from_lds_b128((AS1 v4i*)g, (AS3 v4i*)l, 0, 0);
#else
  unsigned loff = (unsigned)(uintptr_t)(AS3 void*)l;
  unsigned long long ga = (unsigned long long)(uintptr_t)g;
  asm volatile("global_store_async_from_lds_b128 %0, %1, off"
               :: "v"(ga), "v"(loff) : "memory");
#endif
}

__device__ __forceinline__ void wait_async0() {
#if __has_builtin(__builtin_amdgcn_s_wait_asynccnt)
  __builtin_amdgcn_s_wait_asynccnt(0);
#else
  asm volatile("s_wait_asynccnt 0" ::: "memory");
#endif
}

// ---------------- per-row physics -----------------------------------------
__device__ __forceinline__ float mec_torque(float wv, float md) {
  float p = md * wv;
  float isd = p > 0.f ? 1.f : (p < 0.f ? 0.f : 0.5f);  // (sign(p)+1)/2
  return F_T_STALL * (1.f - fabsf(wv) * isd * F_INV_WF) * md;
}

__device__ __forceinline__ void mec_compute(float th, float avx, float avy,
                                            float avz, float md0, float md1,
                                            float md2, float md3, float o[6]) {
  float sn, cs;
  __sincosf(th, &sn, &cs);
  // a2l = rot(-theta): [cos, sin; -sin, cos]
  float lvx = fmaf(cs, avx, sn * avy);
  float lvy = fmaf(cs, avy, -sn * avx);
  // wheel_vel = LV2WV @ local_vel
  float u = lvx * F_INV_R, v = lvy * F_INV_R, w = F_AR * avz;
  float w0 = u - v - w, w1 = u + v + w, w2 = u + v - w, w3 = u - v + w;
  float t0 = mec_torque(w0, md0), t1 = mec_torque(w1, md1);
  float t2 = mec_torque(w2, md2), t3 = mec_torque(w3, md3);
  // local_accel = WT2LA @ torque
  float lax = (t0 + t1 + t2 + t3) * F_K1;
  float lay = (-t0 + t1 + t2 - t3) * F_K1;
  float law = (-t0 + t1 - t2 + t3) * F_K2;
  // l2a = rot(theta): [cos, -sin; sin, cos]
  o[0] = avx;
  o[1] = avy;
  o[2] = avz;
  o[3] = fmaf(cs, lax, -sn * lay);
  o[4] = fmaf(sn, lax, cs * lay);
  o[5] = law;
}

// ---------------- kernel ---------------------------------------------------
#define TPB 256          // 8 waves (wave32)
#define SPT 2            // states per thread
#define TILE (TPB * SPT) // 512 rows / block
#define NF4 (TILE * 6 / 4) // 768 float4 = 12 KB LDS tile

__global__ void __launch_bounds__(TPB)
mecanum_kernel(const float* __restrict__ state, const float* __restrict__ cd,
               float* __restrict__ out, int n) {
  __shared__ float4 smem[NF4];
  const int t = threadIdx.x;
  const long long tileStart = (long long)blockIdx.x * TILE;

  // motor_duty = CD2MD @ control_duty (batch-uniform, 6 ops)
  const float c0 = cd[0], c1 = cd[1], c2 = cd[2];
  const float md0 = c0 - c1 - c2;
  const float md1 = c0 + c1 + c2;
  const float md2 = c0 + c1 - c2;
  const float md3 = c0 - c1 + c2;

  if (tileStart + TILE <= (long long)n) {
    // ---- coalesced async copy: global tile -> LDS (3 x b128 per thread)
    const float4* gin = (const float4*)(state + tileStart * 6);
#pragma unroll
    for (int k = 0; k < 3; ++k) {
      int idx = t + k * TPB;
      async_load_b128(gin + idx, smem + idx);
    }
    wait_async0();       // this wave's async loads done
    __syncthreads();     // all waves' tiles visible in LDS

    float* sf = (float*)smem;
    float res[SPT][6];
#pragma unroll
    for (int k = 0; k < SPT; ++k) {
      int s = t + k * TPB;                 // stride-TPB: LDS conflict-free
      const float* row = sf + s * 6;
      mec_compute(row[2], row[3], row[4], row[5], md0, md1, md2, md3, res[k]);
    }
    // each thread overwrites exactly the rows it read -> no barrier needed here
#pragma unroll
    for (int k = 0; k < SPT; ++k) {
      int s = t + k * TPB;
      float* row = sf + s * 6;
#pragma unroll
      for (int i = 0; i < 6; ++i) row[i] = res[k][i];
    }
    __syncthreads();     // whole tile written before async store

    // ---- coalesced async copy: LDS -> global tile
    float4* gout = (float4*)(out + tileStart * 6);
#pragma unroll
    for (int k = 0; k < 3; ++k) {
      int idx = t + k * TPB;
      async_store_b128(gout + idx, smem + idx);
    }
    wait_async0();       // drain before LDS is released at endpgm
  } else {
    // tail tile (N % TILE rows): direct guarded path
#pragma unroll
    for (int k = 0; k < SPT; ++k) {
      long long row = tileStart + t + k * TPB;
      if (row < (long long)n) {
        const float* p = state + row * 6;
        float res[6];
        mec_compute(p[2], p[3], p[4], p[5], md0, md1, md2, md3, res);
        float* q = out + row * 6;
#pragma unroll
        for (int i = 0; i < 6; ++i) q[i] = res[i];
      }
    }
  }
}

// ---------------- launch ---------------------------------------------------
extern "C" void kernel_launch(void* const* d_in, const int* in_sizes, int n_in,
                              void* d_out, int out_size, void* d_ws,
                              size_t ws_size, hipStream_t stream) {
  (void)n_in; (void)d_ws; (void)ws_size; (void)out_size;
  // inputs (setup_inputs order): 0 = t (unused), 1 = state[B,6], 2 = control_duty[3]
  const float* state = (const float*)d_in[1];
  const float* cd    = (const float*)d_in[2];
  float* out         = (float*)d_out;
  const int n = in_sizes[1] / 6;
  const int blocks = (n + TILE - 1) / TILE;
  mecanum_kernel<<<blocks, TPB, 0, stream>>>(state, cd, out, n);
}